// TimeHomogeneousSDE_34256659153453
// MI455X (gfx1250) — compile-verified
//
#include <hip/hip_runtime.h>

typedef float v2f __attribute__((ext_vector_type(2)));
typedef float v8f __attribute__((ext_vector_type(8)));
typedef unsigned int u32x4 __attribute__((ext_vector_type(4)));
typedef int i32x4 __attribute__((ext_vector_type(4)));
typedef int i32x8 __attribute__((ext_vector_type(8)));

#define BATCH 1024
#define NN    128   // nodes == reduction dim
#define HH    128
#define DD    8

#define HAVE_TDM __has_builtin(__builtin_amdgcn_tensor_load_to_lds)

#if __has_builtin(__builtin_amdgcn_s_wait_tensorcnt)
#define WAIT_TENSORCNT(n) __builtin_amdgcn_s_wait_tensorcnt((short)(n))
#else
#define WAIT_TENSORCNT(n) asm volatile("s_wait_tensorcnt %0" ::"n"(n) : "memory")
#endif

__device__ __forceinline__ float elu1(float v) {
    // jax.nn.elu, alpha=1: x>0 ? x : expm1(x)
    return v > 0.0f ? v : expm1f(v);
}

#if HAVE_TDM
// One TDM op moves a [128 rows x 16 floats] weight tile (8KB) Global->LDS.
// D# group0: count=1 | lds_addr | global_addr(57b) | type=2 ("image")
// D# group1: data_size=4B, tensor_dim0=16, tensor_dim1=128, tile_dim0=16,
//            tile_dim1=128, tensor_dim0_stride=strideElems, no padding/multicast.
__device__ __forceinline__ void tdm_load_tile(const float* gbase, unsigned ldsOff,
                                              int strideElems) {
    const unsigned long long ga = (unsigned long long)gbase;
    u32x4 g0;
    g0.x = 1u;                                            // count=1, user descriptor
    g0.y = ldsOff;                                        // LDS byte address
    g0.z = (unsigned)(ga & 0xffffffffu);                  // global_addr[31:0]
    g0.w = (unsigned)((ga >> 32) & 0x01ffffffu) | (2u << 30); // addr[56:32] | type=2
    i32x8 g1;
    g1[0] = 0x00020000;       // data_size=2 (4B); mask/flags/pad = 0
    g1[1] = (16 << 16);       // tensor_dim0[15:0]  = 16 (bits 63:48)
    g1[2] = (128 << 16);      // tensor_dim1[15:0]  = 128 (bits 95:80)
    g1[3] = (16 << 16);       // tile_dim0 = 16 (bits 127:112)
    g1[4] = 128;              // tile_dim1 = 128 (bits 143:128)
    g1[5] = strideElems;      // tensor_dim0_stride[31:0]
    g1[6] = 0;
    g1[7] = 0;
    const i32x4 z4 = {0, 0, 0, 0};
#if defined(__clang_major__) && (__clang_major__ >= 23)
    const i32x8 z8 = {0, 0, 0, 0, 0, 0, 0, 0};
    __builtin_amdgcn_tensor_load_to_lds(g0, g1, z4, z4, z8, 0);
#else
    __builtin_amdgcn_tensor_load_to_lds(g0, g1, z4, z4, 0);
#endif
}
#endif

// Stage one [128 x 16] f32 weight tile into LDS (dst), rows at strideElems.
__device__ __forceinline__ void stage_tile(float* dst, const float* base,
                                           int strideElems, int tid) {
#if HAVE_TDM
    if ((tid >> 5) == 0)   // one TDM issue per block (wave 0)
        tdm_load_tile(base, (unsigned)(unsigned long long)dst, strideElems);
#else
    // Cooperative fallback: 128 threads x 4 x b128 (load + ds_store).
    const int pr = tid >> 2;          // row 0..31
    const int pc = (tid & 3) * 4;     // col chunk 0,4,8,12
    #pragma unroll
    for (int j = 0; j < 4; ++j) {
        const int r = pr + 32 * j;
        *(float4*)(dst + r * 16 + pc) = *(const float4*)(base + r * strideElems + pc);
    }
#endif
}

__global__ __launch_bounds__(128, 1)
void sde_drift_diff_kernel(const float* __restrict__ x,
                           const float* __restrict__ fw,   // [N,N,H]
                           const float* __restrict__ gw,   // [N,N,H,D]
                           const float* __restrict__ Wf,   // [N,H]
                           const float* __restrict__ bf,   // [N]
                           const float* __restrict__ Wg,   // [N,H]
                           const float* __restrict__ bg,   // [N]
                           float* __restrict__ f_out,      // [B,N]
                           float* __restrict__ g_out)      // [B,N,D]
{
    __shared__ float wtile[3][128 * 16];                   // triple-buffered 8KB tiles

    const int tid  = threadIdx.x;
    const int lane = tid & 31;
    const int wib  = tid >> 5;                             // wave in block 0..3
    const int node = blockIdx.x & (NN - 1);                // block = (node, 64-row slab)
    const int b0   = (blockIdx.x >> 7) * 64 + wib * 16;    // 16-row tile per wave
    const int half = lane >> 4;
    const int l15  = lane & 15;

    const float* fwi = fw + node * (NN * HH);
    const float* gwi = gw + node * (NN * HH * DD);

    // ---- A fragments (x), f32 16x4 layout: lane<16 -> K=4k+{0,1}, lane>=16 -> K=4k+{2,3}
    v2f a[32];
    {
        const float* xrow = x + (b0 + l15) * NN + half * 2;
        #pragma unroll
        for (int kk = 0; kk < 32; ++kk)
            a[kk] = *(const v2f*)(xrow + 4 * kk);
    }

    const float bfv = bf[node];
    const float bgv = bg[node];

    v8f facc = {};
    v8f gacc = {};

    // Prologue: tile 0 (first f tile)
    stage_tile(&wtile[0][0], fwi, HH, tid);

    // 72 column tiles: 0..7 = f panel (H=128), 8..71 = g panel (H*D=1024)
    for (int t = 0; t < 72; ++t) {
        if (t + 1 < 72) {
            const int tn = t + 1;
            const float* base = (tn < 8) ? (fwi + tn * 16) : (gwi + (tn - 8) * 16);
            const int stride  = (tn < 8) ? HH : (HH * DD);
            stage_tile(&wtile[tn % 3][0], base, stride, tid);
#if HAVE_TDM
            if (wib == 0) WAIT_TENSORCNT(1);   // tile t landed; t+1 still in flight
#endif
        } else {
#if HAVE_TDM
            if (wib == 0) WAIT_TENSORCNT(0);
#endif
        }
        __syncthreads();   // all waves see tile t in LDS; frees buffer (t+2)%3 writers

        // ---- 16x16 fp32 WMMA chain over K=128 from LDS.
        // Chunked: preload 8 B-fragments (8x ds_load_2addr) ahead of 8 WMMAs so
        // the scheduler overlaps LDS latency with the matrix pipe within a wave.
        const float* wl = &wtile[t % 3][0];
        v8f acc = {};
        #pragma unroll
        for (int kc = 0; kc < 4; ++kc) {
            v2f bfr[8];
            #pragma unroll
            for (int j = 0; j < 8; ++j) {
                const int k = 4 * (kc * 8 + j) + half * 2;  // half -> K pair, lane -> col
                bfr[j].x = wl[(k    ) * 16 + l15];
                bfr[j].y = wl[(k + 1) * 16 + l15];
            }
            #pragma unroll
            for (int j = 0; j < 8; ++j)
                acc = __builtin_amdgcn_wmma_f32_16x16x4_f32(
                        false, a[kc * 8 + j], false, bfr[j], (short)0, acc, false, false);
        }

        // ---- fused epilogue (uniform branch)
        if (t < 8) {
            const float wfv = Wf[node * HH + t * 16 + l15];
            #pragma unroll
            for (int r = 0; r < 8; ++r)
                facc[r] += elu1(acc[r]) * wfv;          // C/D: VGPR r -> rows r, r+8
        } else {
            const int tg = t - 8;                        // col = tg*16+l15 = h*8+d
            const float wgv = Wg[node * HH + 2 * tg + (l15 >> 3)];
            #pragma unroll
            for (int r = 0; r < 8; ++r)
                gacc[r] += elu1(acc[r]) * wgv;
        }
    }

    // ---- f_out[b,i]: reduce 16 columns per row (xor masks stay inside 16-lane halves)
    #pragma unroll
    for (int r = 0; r < 8; ++r) {
        float v = facc[r];
        v += __shfl_xor(v, 1);
        v += __shfl_xor(v, 2);
        v += __shfl_xor(v, 4);
        v += __shfl_xor(v, 8);
        if (l15 == 0) {
            const int row = b0 + half * 8 + r;
            f_out[row * NN + node] = v + bfv;
        }
    }
    // ---- g_out[b,i,d]: fold lanes n and n+8 (same d), lanes 0..7 / 16..23 store
    #pragma unroll
    for (int r = 0; r < 8; ++r) {
        float v = gacc[r] + __shfl_xor(gacc[r], 8);
        if (l15 < 8) {
            const int row = b0 + half * 8 + r;
            g_out[(row * NN + node) * DD + l15] = v + bgv;
        }
    }
}

extern "C" void kernel_launch(void* const* d_in, const int* in_sizes, int n_in,
                              void* d_out, int out_size, void* d_ws, size_t ws_size,
                              hipStream_t stream) {
    const float* x  = (const float*)d_in[0];
    const float* fw = (const float*)d_in[1];
    const float* gw = (const float*)d_in[2];
    const float* Wf = (const float*)d_in[3];
    const float* bf = (const float*)d_in[4];
    const float* Wg = (const float*)d_in[5];
    const float* bg = (const float*)d_in[6];

    float* f_out = (float*)d_out;                 // [B,N] first in flat tuple order
    float* g_out = f_out + BATCH * NN;            // [B,N,D] follows

    // 2048 blocks = 128 nodes x 16 slabs(64 rows); 4 waves share weights via LDS+TDM.
    sde_drift_diff_kernel<<<dim3(2048), dim3(128), 0, stream>>>(
        x, fw, gw, Wf, bf, Wg, bg, f_out, g_out);
}